// ContextEmbedding_35012573397647
// MI455X (gfx1250) — compile-verified
//
#include <hip/hip_runtime.h>

#define NUM_SPECIAL    8
#define NUM_CONTEXT    16
#define SPECIAL_OFFSET 72      // 52 + 20 bet bins
#define D_MODEL        256
#define LN_EPS         1e-5f

typedef float v2f __attribute__((ext_vector_type(2)));
typedef float v8f __attribute__((ext_vector_type(8)));

__global__ __launch_bounds__(256)
void ContextEmbedding_35012573397647_kernel(
    const int*   __restrict__ token_ids,   // [N]
    const float* __restrict__ ctx_feat,    // [N,16]
    const float* __restrict__ emb_table,   // [8,256]
    const float* __restrict__ W_cls,       // [3,256]
    const float* __restrict__ b_cls,       // [256]
    const float* __restrict__ g_cls,       // [256]
    const float* __restrict__ beta_cls,    // [256]
    const float* __restrict__ W_ctx,       // [16,256]
    const float* __restrict__ b_ctx,       // [256]
    const float* __restrict__ g_ctx,       // [256]
    const float* __restrict__ beta_ctx,    // [256]
    float*       __restrict__ out,         // [N,256]
    int n_tokens)
{
    __shared__ float s_feat[16][NUM_CONTEXT];   // 16 tokens x 16 features
    __shared__ int   s_ids[16];
    __shared__ float s_hctx[16][D_MODEL];       // pre-LN ctx MLP activations
    __shared__ float s_hcls[16][D_MODEL];       // pre-LN cls MLP activations

    const int tid  = threadIdx.x;
    const int wave = tid >> 5;
    const int lane = tid & 31;
    const int base = blockIdx.x * 16;           // first token of this tile

    // ---------------- Phase 1: stage ids + features into LDS ----------------
    if (tid < 16) {
        int g = base + tid;
        s_ids[tid] = (g < n_tokens) ? token_ids[g] : 0;
    }
    {
        int r = tid >> 4;                       // token row 0..15
        int f = tid & 15;                       // feature 0..15
        int g = base + r;
        s_feat[r][f] = (g < n_tokens) ? ctx_feat[g * NUM_CONTEXT + f] : 0.0f;
    }
    __syncthreads();

    // ---------------- Phase 2: dense WMMA GEMMs (f32, 16x16x4) --------------
    // A layout (16x4 f32, ISA 7.12.2): lanes 0-15 -> M=lane, regs hold K={0,1};
    //                                  lanes 16-31 -> same M, K={2,3}.
    // B layout mirrors with lane = N.
    const int M    = lane & 15;
    const int half = lane >> 4;                 // 0: K lo pair, 1: K hi pair

    v2f a_ctx[4];
    #pragma unroll
    for (int kk = 0; kk < 4; ++kk) {
        int k0 = 4 * kk + 2 * half;
        a_ctx[kk].x = s_feat[M][k0];
        a_ctx[kk].y = s_feat[M][k0 + 1];
    }
    v2f a_cls;                                   // K=3 padded to 4 with zeros
    a_cls.x = (half == 0) ? s_feat[M][0] : s_feat[M][2];
    a_cls.y = (half == 0) ? s_feat[M][1] : 0.0f;

    #pragma unroll
    for (int t = 0; t < 2; ++t) {
        const int n0  = (wave * 2 + t) * 16;     // this wave's 16-col tile
        const int col = n0 + M;

        v8f c_ctx = {};
        #pragma unroll
        for (int kk = 0; kk < 4; ++kk) {
            int k0 = 4 * kk + 2 * half;
            v2f b;
            b.x = W_ctx[k0 * D_MODEL + col];
            b.y = W_ctx[(k0 + 1) * D_MODEL + col];
            c_ctx = __builtin_amdgcn_wmma_f32_16x16x4_f32(
                false, a_ctx[kk], false, b, (short)0, c_ctx, false, false);
        }

        v2f bk;                                  // W_cls rows 0..2, row 3 = 0
        bk.x = (half == 0) ? W_cls[0 * D_MODEL + col] : W_cls[2 * D_MODEL + col];
        bk.y = (half == 0) ? W_cls[1 * D_MODEL + col] : 0.0f;
        v8f c_cls = {};
        c_cls = __builtin_amdgcn_wmma_f32_16x16x4_f32(
            false, a_cls, false, bk, (short)0, c_cls, false, false);

        // C/D layout: VGPR v holds row v + 8*half, col = n0 + M
        const float biasc = b_ctx[col];
        const float biask = b_cls[col];
        #pragma unroll
        for (int v = 0; v < 8; ++v) {
            int row = v + 8 * half;
            s_hctx[row][col] = c_ctx[v] + biasc;
            s_hcls[row][col] = c_cls[v] + biask;
        }
    }
    __syncthreads();

    // -------- Phase 3: per-token LayerNorm + ReLU + mask + gather + store ---
    #pragma unroll
    for (int rr = 0; rr < 2; ++rr) {
        const int row  = wave * 2 + rr;          // token row handled by wave
        const int gtok = base + row;
        const int id   = s_ids[row];

        // partial sums: lane owns cols [4*lane,4*lane+4) and +128
        float s1 = 0.f, s2 = 0.f, q1 = 0.f, q2 = 0.f;
        #pragma unroll
        for (int j = 0; j < 2; ++j) {
            int c0 = 4 * lane + 128 * j;
            #pragma unroll
            for (int u = 0; u < 4; ++u) {
                float hc = s_hctx[row][c0 + u];
                float hk = s_hcls[row][c0 + u];
                s1 += hc; s2 += hc * hc;
                q1 += hk; q2 += hk * hk;
            }
        }
        // wave32 butterfly reduction
        #pragma unroll
        for (int off = 16; off >= 1; off >>= 1) {
            s1 += __shfl_xor(s1, off, 32);
            s2 += __shfl_xor(s2, off, 32);
            q1 += __shfl_xor(q1, off, 32);
            q2 += __shfl_xor(q2, off, 32);
        }
        const float inv  = 1.0f / (float)D_MODEL;
        const float mu_c = s1 * inv;
        const float rs_c = rsqrtf(fmaxf(s2 * inv - mu_c * mu_c, 0.0f) + LN_EPS);
        const float mu_k = q1 * inv;
        const float rs_k = rsqrtf(fmaxf(q2 * inv - mu_k * mu_k, 0.0f) + LN_EPS);

        const int  sid        = id - SPECIAL_OFFSET;
        const bool is_special = (sid >= 0) && (sid < NUM_SPECIAL);
        const bool is_cls     = (sid == 0);       // Special.CLS
        const bool is_ctx     = (sid == 1);       // Special.CONTEXT
        const float* erow = emb_table + (is_special ? sid : 0) * D_MODEL;

        #pragma unroll
        for (int j = 0; j < 2; ++j) {
            int c0 = 4 * lane + 128 * j;
            float4 o;
            float* op = &o.x;
            #pragma unroll
            for (int u = 0; u < 4; ++u) {
                int c = c0 + u;
                float e = is_special ? erow[c] : 0.0f;
                if (is_cls) {
                    float h = (s_hcls[row][c] - mu_k) * rs_k * g_cls[c] + beta_cls[c];
                    e += fmaxf(h, 0.0f);
                }
                if (is_ctx) {
                    float h = (s_hctx[row][c] - mu_c) * rs_c * g_ctx[c] + beta_ctx[c];
                    e += fmaxf(h, 0.0f);
                }
                op[u] = e;
            }
            if (gtok < n_tokens)
                *(float4*)(out + (size_t)gtok * D_MODEL + c0) = o;  // B128 store
        }
    }
}

extern "C" void kernel_launch(void* const* d_in, const int* in_sizes, int n_in,
                              void* d_out, int out_size, void* d_ws, size_t ws_size,
                              hipStream_t stream) {
    const int*   token_ids = (const int*)  d_in[0];
    const float* ctx_feat  = (const float*)d_in[1];
    const float* emb_table = (const float*)d_in[2];
    const float* W_cls     = (const float*)d_in[3];
    const float* b_cls     = (const float*)d_in[4];
    const float* g_cls     = (const float*)d_in[5];
    const float* beta_cls  = (const float*)d_in[6];
    const float* W_ctx     = (const float*)d_in[7];
    const float* b_ctx     = (const float*)d_in[8];
    const float* g_ctx     = (const float*)d_in[9];
    const float* beta_ctx  = (const float*)d_in[10];
    float*       out       = (float*)d_out;

    const int n_tokens = in_sizes[0];            // B*S = 131072
    const int n_blocks = (n_tokens + 15) / 16;   // 16 tokens per block

    ContextEmbedding_35012573397647_kernel<<<n_blocks, 256, 0, stream>>>(
        token_ids, ctx_feat, emb_table,
        W_cls, b_cls, g_cls, beta_cls,
        W_ctx, b_ctx, g_ctx, beta_ctx,
        out, n_tokens);
}